// Model6GruLinear_80745385165139
// MI455X (gfx1250) — compile-verified
//
#include <hip/hip_runtime.h>

// Model: GRU(I=8,H=6) over [B=4096, T=512] + Linear(6->4).
// Pass 1 (parallel, WMMA): xg[t*B+b][0..17] = x[b,t,:] @ W_ih^T + b_ih
//   M = B*T = 2,097,152 rows, K=8, N=18 -> V_WMMA_F32_16X16X4_F32 tiles.
// Pass 2 (serial scan, VALU): per-lane GRU recurrence + fused output linear.
// Rationale: the scan is inherently serial (nonlinear), parallelism capped at
// B=4096 lanes; hoisting the input GEMM off the serial path halves per-step
// work, and that GEMM is the natural WMMA target. Traffic ~= 64MB x read +
// 160MB xg write + 160MB xg read + 32MB out ~ 17us at 23.3 TB/s; the serial
// recurrence (512 dependent steps of ~160 VALU ops) dominates regardless.

#define B_SZ   4096
#define LOG2_B 12
#define T_SZ   512
#define I_SZ   8
#define H_SZ   6
#define G3     18          // 3*H
#define XG_STRIDE 20       // pad 18 -> 20 floats so rows stay 16B-aligned
#define O_SZ   4
#define M_ROWS (B_SZ * T_SZ)

typedef __attribute__((ext_vector_type(2))) float v2f;
typedef __attribute__((ext_vector_type(8))) float v8f;

// ---------------------------------------------------------------------------
// Pass 1: xg = x @ W_ih^T + b_ih via V_WMMA_F32_16X16X4_F32.
// One wave per 16-row tile of xg (row index m = t*B + b so pass 2 is
// coalesced). A-layout (ISA 7.12.2): lanes 0-15 hold M=0..15 K={0,1} in
// {v0,v1}; lanes 16-31 hold K={2,3}. B-layout mirrors with N per lane.
// C-layout: VGPR r = row (r or r+8 by half-wave), lane&15 = column.
// ---------------------------------------------------------------------------
__global__ __launch_bounds__(512)
void Model6Gru_xg_wmma(const float* __restrict__ x,
                       const float* __restrict__ W_ih,
                       const float* __restrict__ b_ih,
                       float* __restrict__ xg) {
  const int lane = threadIdx.x & 31;
  const int wave = threadIdx.x >> 5;
  const int tile = blockIdx.x * (blockDim.x >> 5) + wave;
  const int m0   = tile << 4;          // first xg row of this 16-row tile
  const int half = lane >> 4;          // 0: K base 0 ; 1: K base 2
  const int nr   = lane & 15;          // A: row-in-tile ; B/C: column

  // ---- A operands: x row for xg-row m (m = t*B + b  ->  x[b*T + t]) ----
  const int m    = m0 + nr;
  const int brow = m & (B_SZ - 1);
  const int trow = m >> LOG2_B;
  const float* xrow = x + ((size_t)brow * T_SZ + trow) * I_SZ;
  const int k0 = half * 2;
  v2f a0, a1;                          // K=0..3 and K=4..7 slices
  a0.x = xrow[k0];     a0.y = xrow[k0 + 1];
  a1.x = xrow[4 + k0]; a1.y = xrow[4 + k0 + 1];

  // ---- B operands: W_ih^T columns. Main tile = gates 0..15; second tile
  // carries gates 16,17 in columns 0,1 (other columns clamped in-bounds,
  // results discarded). No divergence before the WMMAs. ----
  v2f b0, b1, b2, b3;
  b0.x = W_ih[nr * I_SZ + k0];         b0.y = W_ih[nr * I_SZ + k0 + 1];
  b1.x = W_ih[nr * I_SZ + 4 + k0];     b1.y = W_ih[nr * I_SZ + 4 + k0 + 1];
  const int g2 = (nr < 2) ? (16 + nr) : 17;   // clamp: stay in-bounds
  b2.x = W_ih[g2 * I_SZ + k0];         b2.y = W_ih[g2 * I_SZ + k0 + 1];
  b3.x = W_ih[g2 * I_SZ + 4 + k0];     b3.y = W_ih[g2 * I_SZ + 4 + k0 + 1];

  // ---- C init with bias (bias is per-column == per-gate) ----
  const float cb  = b_ih[nr];
  const float cb2 = (nr < 2) ? b_ih[16 + nr] : 0.0f;
  v8f c  = {cb,  cb,  cb,  cb,  cb,  cb,  cb,  cb};
  v8f c2 = {cb2, cb2, cb2, cb2, cb2, cb2, cb2, cb2};

  // ---- 4x v_wmma_f32_16x16x4_f32 (K=8 -> 2 chained per N-tile) ----
  c  = __builtin_amdgcn_wmma_f32_16x16x4_f32(false, a0, false, b0, (short)0, c,  false, false);
  c  = __builtin_amdgcn_wmma_f32_16x16x4_f32(false, a1, false, b1, (short)0, c,  false, false);
  c2 = __builtin_amdgcn_wmma_f32_16x16x4_f32(false, a0, false, b2, (short)0, c2, false, false);
  c2 = __builtin_amdgcn_wmma_f32_16x16x4_f32(false, a1, false, b3, (short)0, c2, false, false);

  // ---- Store: lane (half,nr) holds column nr of rows m0+half*8+r ----
  const int rbase = m0 + half * 8;
#pragma unroll
  for (int r = 0; r < 8; ++r)
    xg[(size_t)(rbase + r) * XG_STRIDE + nr] = c[r];
  if (nr < 2) {
#pragma unroll
    for (int r = 0; r < 8; ++r)
      xg[(size_t)(rbase + r) * XG_STRIDE + 16 + nr] = c2[r];
  }
}

// ---------------------------------------------------------------------------
// Pass 2: serial GRU scan + fused Linear. One lane per batch row; all
// weights live in VGPRs (reused 512x). xg laid out [t][b] -> coalesced.
// ---------------------------------------------------------------------------
__device__ __forceinline__ float fsigmoid(float v) {
  return __builtin_amdgcn_rcpf(1.0f + __expf(-v));       // v_exp + v_rcp
}
__device__ __forceinline__ float ftanh(float v) {
  // tanh(v) = 1 - 2/(1 + e^{2v}); saturates correctly at +/-inf.
  return 1.0f - 2.0f * __builtin_amdgcn_rcpf(1.0f + __expf(2.0f * v));
}

__global__ __launch_bounds__(256)
void Model6Gru_rec(const float* __restrict__ xg,
                   const float* __restrict__ W_hh,
                   const float* __restrict__ b_hh,
                   const float* __restrict__ W_lin,
                   const float* __restrict__ b_lin,
                   float* __restrict__ out) {
  const int b = blockIdx.x * blockDim.x + threadIdx.x;   // 0..4095

  float whh[G3][H_SZ], bhh[G3], wlin[O_SZ][H_SZ], blin[O_SZ];
#pragma unroll
  for (int g = 0; g < G3; ++g) {
    bhh[g] = b_hh[g];
#pragma unroll
    for (int k = 0; k < H_SZ; ++k) whh[g][k] = W_hh[g * H_SZ + k];
  }
#pragma unroll
  for (int o = 0; o < O_SZ; ++o) {
    blin[o] = b_lin[o];
#pragma unroll
    for (int k = 0; k < H_SZ; ++k) wlin[o][k] = W_lin[o * H_SZ + k];
  }

  float h[H_SZ] = {0.f, 0.f, 0.f, 0.f, 0.f, 0.f};
  float* op = out + (size_t)b * T_SZ * O_SZ;

#pragma unroll 2
  for (int t = 0; t < T_SZ; ++t) {
    const float* xr = xg + ((size_t)t * B_SZ + b) * XG_STRIDE;
    if (t + 4 < T_SZ)                                    // hide HBM latency
      __builtin_prefetch(xr + (size_t)4 * B_SZ * XG_STRIDE, 0, 0);

    float4 l0 = *(const float4*)(xr);
    float4 l1 = *(const float4*)(xr + 4);
    float4 l2 = *(const float4*)(xr + 8);
    float4 l3 = *(const float4*)(xr + 12);
    float2 l4 = *(const float2*)(xr + 16);
    const float xv[G3] = {l0.x, l0.y, l0.z, l0.w, l1.x, l1.y,
                          l1.z, l1.w, l2.x, l2.y, l2.z, l2.w,
                          l3.x, l3.y, l3.z, l3.w, l4.x, l4.y};

    float gh[G3];
#pragma unroll
    for (int g = 0; g < G3; ++g) {
      float acc = bhh[g];
#pragma unroll
      for (int k = 0; k < H_SZ; ++k) acc = fmaf(whh[g][k], h[k], acc);
      gh[g] = acc;
    }

#pragma unroll
    for (int i = 0; i < H_SZ; ++i) {
      const float r = fsigmoid(xv[i]      + gh[i]);
      const float z = fsigmoid(xv[6 + i]  + gh[6 + i]);
      const float n = ftanh   (xv[12 + i] + r * gh[12 + i]);
      h[i] = (1.0f - z) * n + z * h[i];
    }

    float4 o;
    {
      float a0 = blin[0], a1 = blin[1], a2 = blin[2], a3 = blin[3];
#pragma unroll
      for (int k = 0; k < H_SZ; ++k) {
        a0 = fmaf(wlin[0][k], h[k], a0);
        a1 = fmaf(wlin[1][k], h[k], a1);
        a2 = fmaf(wlin[2][k], h[k], a2);
        a3 = fmaf(wlin[3][k], h[k], a3);
      }
      o.x = a0; o.y = a1; o.z = a2; o.w = a3;
    }
    *(float4*)(op + (size_t)t * O_SZ) = o;
  }
}

// ---------------------------------------------------------------------------
extern "C" void kernel_launch(void* const* d_in, const int* in_sizes, int n_in,
                              void* d_out, int out_size, void* d_ws, size_t ws_size,
                              hipStream_t stream) {
  const float* x     = (const float*)d_in[0];
  const float* W_ih  = (const float*)d_in[1];
  const float* W_hh  = (const float*)d_in[2];
  const float* b_ih  = (const float*)d_in[3];
  const float* b_hh  = (const float*)d_in[4];
  const float* W_lin = (const float*)d_in[5];
  const float* b_lin = (const float*)d_in[6];
  float* out = (float*)d_out;
  float* xg  = (float*)d_ws;   // needs M_ROWS * 20 * 4 = 160 MiB

  // Pass 1: 131072 wave-tiles, 16 waves (512 thr) per block -> 8192 blocks.
  const int waves_per_blk = 512 / 32;
  const int tiles = M_ROWS / 16;
  Model6Gru_xg_wmma<<<dim3(tiles / waves_per_blk), dim3(512), 0, stream>>>(
      x, W_ih, b_ih, xg);

  // Pass 2: one lane per batch row.
  Model6Gru_rec<<<dim3(B_SZ / 256), dim3(256), 0, stream>>>(
      xg, W_hh, b_hh, W_lin, b_lin, out);
}